// KernelAttnCoef_17351667876262
// MI455X (gfx1250) — compile-verified
//
#include <hip/hip_runtime.h>

// ---------------------------------------------------------------------------
// Problem constants (from the reference)
// ---------------------------------------------------------------------------
constexpr int kB      = 2;
constexpr int kN      = 4096;                    // nodes per batch
constexpr int kE      = 16384;                   // edges per batch
constexpr int kH      = 8;
constexpr int kDQK    = 16;
constexpr int kDV     = 128;
constexpr int kNV     = kB * kE;                 // 32768 edges total
constexpr int kNTOT   = kB * kN;                 // 8192 nodes total
constexpr int kCH     = kDQK * kDV + kH * kDQK;  // 2176 accum floats / node
constexpr int kESHIFT = 14;                      // log2(kE)

typedef float v2f __attribute__((ext_vector_type(2)));
typedef float v8f __attribute__((ext_vector_type(8)));

// D(16x16 f32) = A(16x4 f32) * B(4x16 f32) + C   -- f32 WMMA, exact products
static __device__ __forceinline__ v8f wmma_f32_16x16x4(v2f a, v2f b, v8f c) {
    return __builtin_amdgcn_wmma_f32_16x16x4_f32(
        /*neg_a=*/false, a, /*neg_b=*/false, b,
        /*c_mod=*/(short)0, c, /*reuse_a=*/false, /*reuse_b=*/false);
}

// ---------------------------------------------------------------------------
// Kernel 0: zero the 71.3 MB node accumulator (L2-resident on MI455X, 192MB L2)
// ---------------------------------------------------------------------------
__global__ void zero_ws_kernel(float4* __restrict__ ws, long long n4) {
    long long i      = (long long)blockIdx.x * blockDim.x + threadIdx.x;
    long long stride = (long long)gridDim.x * blockDim.x;
    for (; i < n4; i += stride) ws[i] = make_float4(0.f, 0.f, 0.f, 0.f);
}

// ---------------------------------------------------------------------------
// Kernel 1: scatter  (one wave32 per edge, 4 waves per block)
//   per head h: D = k_h (16) outer v_h (16) via one K=4 WMMA (K1..3 zero),
//   then 2048 + 128 f32 atomics into the destination node's accumulator.
// ---------------------------------------------------------------------------
__global__ void __launch_bounds__(128)
scatter_kv_kernel(const float* __restrict__ kvals,
                  const float* __restrict__ vvals,
                  const int*   __restrict__ vidx,
                  float*       __restrict__ acc) {
    const int lane = threadIdx.x & 31;
    const int wave = threadIdx.x >> 5;
    const int edge = (blockIdx.x << 2) + wave;          // < kNV by grid sizing

    const int b   = edge >> kESHIFT;
    const int row = vidx[edge * 2 + 0] + b * kN;        // k gather node
    const int col = vidx[edge * 2 + 1] + b * kN;        // scatter node

    const float* __restrict__ kg  = kvals + (long long)row  * (kH * kDQK);
    const float* __restrict__ vv  = vvals + (long long)edge * kDV;
    float*       __restrict__ dst = acc   + (long long)col  * kCH;

    const int j  = lane & 15;   // N column (and M row for A)
    const int hi = lane >> 4;   // lane half: selects K=0 (used) vs K=2 (zero)

    #pragma unroll
    for (int h = 0; h < kH; ++h) {
        const float ka = kg[h * 16 + j];   // A[m=j, K=0] = k_h[m]
        const float vb = vv[h * 16 + j];   // B[K=0, n=j] = v_h[n]
        v2f A;  A.x = (hi == 0) ? ka : 0.f;  A.y = 0.f;   // K=2/3 -> 0
        v2f Bm; Bm.x = (hi == 0) ? vb : 0.f; Bm.y = 0.f;
        v8f C = {};
        C = wmma_f32_16x16x4(A, Bm, C);    // D[i][j] = k_h[i]*v_h[j], exact

        #pragma unroll
        for (int r = 0; r < 8; ++r) {
            const int m = r + (hi << 3);   // D row of this VGPR
            // kv laid out as (i, h, j) -> i*128 + h*16 + j
            atomicAdd(dst + m * 128 + h * 16 + j, C[r]);
        }
    }
    // kg_flat tail: acc[2048 + t] += k_g[t]
    #pragma unroll
    for (int t = lane; t < kH * kDQK; t += 32)
        atomicAdd(dst + kDQK * kDV + t, kg[t]);
}

// ---------------------------------------------------------------------------
// Kernel 2: gather  (one wave32 per edge, 4 waves per block)
//   att[h,j] = sum_k Aq[h,k]*KV[k,j], Aq = block-diag expanded q (16x128),
//   done as 32 chained K=4 f32 WMMAs. qk_h via lane partials + shfl reduce.
// ---------------------------------------------------------------------------
__global__ void __launch_bounds__(128)
gather_att_kernel(const float* __restrict__ qvals,
                  const int*   __restrict__ qidx,
                  const float* __restrict__ acc,
                  float*       __restrict__ out) {
    const int lane = threadIdx.x & 31;
    const int wave = threadIdx.x >> 5;
    const int edge = (blockIdx.x << 2) + wave;

    const int b    = edge >> kESHIFT;
    const int qi0  = qidx[edge * 2 + 0];
    const int qi1  = qidx[edge * 2 + 1];
    const int node = qi1 + b * kN;

    const float* __restrict__ g = acc   + (long long)node * kCH;
    const float* __restrict__ q = qvals + (long long)edge * (kH * kDQK);

    const int n   = lane & 15;   // N column; also M row index for A
    const int hi  = lane >> 4;
    const int kk0 = hi << 1;     // lane-half K offset within 4-chunk

    v8f C = {};
    #pragma unroll 4
    for (int c = 0; c < 32; ++c) {          // K chunks of 4 over k=(h',i)
        const int hh = c >> 2;              // head h' of this chunk
        const int i0 = ((c & 3) << 2) + kk0;// within-head i for this lane half
        v2f Bm;                             // B[k,n] = KV[(h',i), n] = g[i*128+h'*16+n]
        Bm.x = g[i0 * 128 + hh * 16 + n];
        Bm.y = g[(i0 + 1) * 128 + hh * 16 + n];
        const float q0 = q[hh * 16 + i0];
        const float q1 = q[hh * 16 + i0 + 1];
        v2f A;                              // A[m,k] = q[k] iff m == h'
        A.x = (n == hh) ? q0 : 0.f;
        A.y = (n == hh) ? q1 : 0.f;
        C = wmma_f32_16x16x4(A, Bm, C);
    }
    // rows 0..7 of C = att[h][j]; rows 8..15 are exactly zero.

    // qk_h = sum_i q[h,i] * ksum[h,i];  lane (h=lane>>2, seg=lane&3) partial
    const int h   = lane >> 2;
    const int seg = lane & 3;
    float p = 0.f;
    #pragma unroll
    for (int i = 0; i < 4; ++i) {
        const int idx = h * 16 + seg * 4 + i;
        p += q[idx] * g[kDQK * kDV + idx];
    }
    p += __shfl_xor(p, 1, 32);
    p += __shfl_xor(p, 2, 32);              // lanes 4h..4h+3 hold qk[h]

    float qk[8];
    #pragma unroll
    for (int r = 0; r < 8; ++r) qk[r] = __shfl(p, 4 * r, 32);

    const bool keep = (qi0 != qi1);         // self-loop mask
    if (hi == 0) {                          // lanes 0..15 hold heads 0..7
        float* __restrict__ o = out + (long long)edge * kDV;
        #pragma unroll
        for (int r = 0; r < 8; ++r) {
            float d = qk[r];
            d = (d == 0.f) ? 1e-5f : d;
            o[r * 16 + n] = keep ? (C[r] / d) : 0.f;
        }
    }
}

// ---------------------------------------------------------------------------
// Launch: zero accumulator -> scatter -> gather. All on `stream`.
// Workspace need: kNTOT * kCH * 4 bytes = 71,303,168 B.
// ---------------------------------------------------------------------------
extern "C" void kernel_launch(void* const* d_in, const int* in_sizes, int n_in,
                              void* d_out, int out_size, void* d_ws, size_t ws_size,
                              hipStream_t stream) {
    const float* qv = (const float*)d_in[0];   // query_values  (B,E,H*DQK)
    const float* kv = (const float*)d_in[1];   // key_values    (B,N,H*DQK)
    const float* vv = (const float*)d_in[2];   // value_values  (B,E,DV)
    const int*   qi = (const int*)  d_in[3];   // query_indices (B,E,2)
    const int*   vi = (const int*)  d_in[4];   // value_indices (B,E,2)
    float*       out = (float*)d_out;
    float*       acc = (float*)d_ws;

    const long long acc_f32 = (long long)kNTOT * kCH;   // 17,825,792 (mult of 4)
    zero_ws_kernel<<<2048, 256, 0, stream>>>((float4*)acc, acc_f32 / 4);
    scatter_kv_kernel<<<kNV / 4, 128, 0, stream>>>(kv, vv, vi, acc);
    gather_att_kernel<<<kNV / 4, 128, 0, stream>>>(qv, qi, acc, out);
}